// Gate_36404142801382
// MI455X (gfx1250) — compile-verified
//
#include <hip/hip_runtime.h>
#include <math.h>

// Problem constants (from reference): x[B,T,D], density = T*0.9
#define B_      32
#define T_      4096
#define D_      384
#define DENSITY 3686
#define KSKIP   410

typedef __attribute__((ext_vector_type(2))) float v2f;
typedef __attribute__((ext_vector_type(8))) float v8f;

// ---------------------------------------------------------------------------
// Kernel 1: logits[b,t] = dot(x[b,t,:], w) + bias   via V_WMMA_F32_16X16X4_F32
// One wave32 handles a 16-token tile. A = 16x4 fp32 tile of x (lane L<16 holds
// row L, K={kk,kk+1}; lane L>=16 holds row L-16, K={kk+2,kk+3} -> float2 load).
// B = 4x16 with w replicated into every column, so every column of the 16x16
// f32 accumulator equals the 16 logits. 96 accumulating WMMAs cover D=384.
// EXEC is all-ones through the WMMA loop (full 256-thread blocks, no branches).
// ---------------------------------------------------------------------------
__global__ __launch_bounds__(256) void logits_kernel(const float* __restrict__ x,
                                                     const float* __restrict__ w,
                                                     const float* __restrict__ bh,
                                                     float* __restrict__ logits) {
  const int lane = threadIdx.x & 31;
  const int wid  = blockIdx.x * (256 / 32) + (threadIdx.x >> 5);   // global wave id
  const int b    = wid >> 8;                 // 256 tiles of 16 tokens per batch row
  const int t0   = (wid & 255) << 4;
  const int m    = lane & 15;                // A-matrix row owned by this lane
  const int koff = (lane >> 4) << 1;         // K sub-offset: 0 for lanes 0-15, 2 for 16-31

  const float* xrow = x + ((size_t)b * T_ + t0 + m) * D_;

  v8f acc = {};
  for (int kk = 0; kk < D_; kk += 4) {
    const float2 av = *(const float2*)(xrow + kk + koff);   // 8B-aligned (even offset)
    v2f a;  a.x = av.x;            a.y = av.y;
    v2f bm; bm.x = w[kk + koff];   bm.y = w[kk + koff + 1]; // broadcast within half-wave
    // 8 args: (neg_a, A, neg_b, B, c_mod, C, reuse_a, reuse_b)
    acc = __builtin_amdgcn_wmma_f32_16x16x4_f32(false, a, false, bm,
                                                (short)0, acc, false, false);
  }

  // C/D layout: lane L, VGPR r holds D[M = r + 8*(L>>4)][N = L&15]; all N equal.
  if ((lane & 15) == 0) {
    const float bias = bh[0];
    float* lg = logits + (size_t)b * T_ + t0 + ((lane >> 4) << 3);
#pragma unroll
    for (int r = 0; r < 8; ++r) lg[r] = acc[r] + bias;
  }
}

// ---------------------------------------------------------------------------
// Kernel 2: per batch row, bitonic sort (value desc, index asc on ties) of all
// 4096 (logit, index) pairs in LDS; then compute softmax weights for the keep
// region [0, DENSITY) and skip region [DENSITY, T). 36 KB LDS (<< 320 KB/WGP).
// ---------------------------------------------------------------------------
__global__ __launch_bounds__(1024) void sort_softmax_kernel(const float* __restrict__ logits,
                                                            int* __restrict__ sidx,
                                                            float* __restrict__ wsor) {
  __shared__ float vals[T_];
  __shared__ int   inds[T_];
  __shared__ float red[1024];
  __shared__ float bc[2];

  const int tid = threadIdx.x;
  const int b   = blockIdx.x;
  const float* lrow = logits + (size_t)b * T_;

  for (int p = tid; p < T_; p += 1024) { vals[p] = lrow[p]; inds[p] = p; }
  __syncthreads();

  // Bitonic sort, globally descending; stable tiebreak = lower index first.
  for (unsigned k = 2; k <= (unsigned)T_; k <<= 1) {
    for (unsigned j = k >> 1; j > 0; j >>= 1) {
      for (unsigned i = tid; i < (unsigned)T_; i += 1024) {
        const unsigned ixj = i ^ j;
        if (ixj > i) {
          const float v0 = vals[i], v1 = vals[ixj];
          const int   i0 = inds[i], i1 = inds[ixj];
          const bool beats = (v1 > v0) || (v1 == v0 && i1 < i0); // ixj outranks i
          const bool desc  = ((i & k) == 0);
          if (desc == beats) {
            vals[i] = v1; vals[ixj] = v0;
            inds[i] = i1; inds[ixj] = i0;
          }
        }
      }
      __syncthreads();
    }
  }

  // Softmax normalizers. Keep max = vals[0]; skip region max = vals[DENSITY]
  // (skip region is descending; reversed it is the ascending top_k(-logits)).
  const float m_keep = vals[0];
  const float m_skip = vals[DENSITY];

  float s = 0.f;
  for (int p = tid; p < DENSITY; p += 1024) s += __expf(vals[p] - m_keep);
  red[tid] = s;
  for (int off = 512; off > 0; off >>= 1) {
    __syncthreads();
    if (tid < off) red[tid] += red[tid + off];
  }
  __syncthreads();
  if (tid == 0) bc[0] = red[0];
  __syncthreads();

  s = 0.f;
  for (int p = DENSITY + tid; p < T_; p += 1024) s += __expf(vals[p] - m_skip);
  red[tid] = s;
  for (int off = 512; off > 0; off >>= 1) {
    __syncthreads();
    if (tid < off) red[tid] += red[tid + off];
  }
  __syncthreads();
  if (tid == 0) bc[1] = red[0];
  __syncthreads();

  const float inv_keep = 1.f / bc[0];
  const float inv_skip = 1.f / bc[1];
  for (int p = tid; p < T_; p += 1024) {
    const float v = vals[p];
    wsor[(size_t)b * T_ + p] = (p < DENSITY) ? __expf(v - m_keep) * inv_keep
                                             : __expf(v - m_skip) * inv_skip;
    sidx[(size_t)b * T_ + p] = inds[p];
  }
}

// ---------------------------------------------------------------------------
// Kernel 3: gather-permute. One 96-thread block copies one 384-float row with
// float4 (1536 B). p < DENSITY -> tokens[p]; else -> skip_tokens[T-1-p]
// (reversed tail of the descending sort = ascending skip order).
// ---------------------------------------------------------------------------
__global__ __launch_bounds__(96) void gather_kernel(const float* __restrict__ x,
                                                    const int* __restrict__ sidx,
                                                    float* __restrict__ out_tok,
                                                    float* __restrict__ out_skip) {
  const int pbl = blockIdx.x;
  const int b   = pbl >> 12;            // T_ == 4096
  const int p   = pbl & (T_ - 1);
  const int src = sidx[(size_t)b * T_ + p];

  const float4* sp = (const float4*)(x + ((size_t)b * T_ + src) * D_);
  float4* dp;
  if (p < DENSITY) {
    dp = (float4*)(out_tok + ((size_t)b * DENSITY + p) * D_);
  } else {
    dp = (float4*)(out_skip + ((size_t)b * KSKIP + (T_ - 1 - p)) * D_);
  }
  dp[threadIdx.x] = sp[threadIdx.x];
}

// ---------------------------------------------------------------------------
// Kernel 4: weighted row reductions over the gathered tokens (coalesced in d,
// weight is a uniform broadcast load per iteration).
// ---------------------------------------------------------------------------
__global__ __launch_bounds__(384) void summary_kernel(const float* __restrict__ tok,
                                                      const float* __restrict__ skiptok,
                                                      const float* __restrict__ wsor,
                                                      float* __restrict__ sum_keep,
                                                      float* __restrict__ sum_skip) {
  const int b = blockIdx.x;
  const int d = threadIdx.x;
  const float* wb = wsor + (size_t)b * T_;
  float acc = 0.f;
  if (blockIdx.y == 0) {
    const float* tb = tok + (size_t)b * DENSITY * D_;
#pragma unroll 4
    for (int j = 0; j < DENSITY; ++j) acc += wb[j] * tb[(size_t)j * D_ + d];
    sum_keep[(size_t)b * D_ + d] = acc;
  } else {
    const float* tb = skiptok + (size_t)b * KSKIP * D_;
#pragma unroll 4
    for (int j = 0; j < KSKIP; ++j) acc += wb[T_ - 1 - j] * tb[(size_t)j * D_ + d];
    sum_skip[(size_t)b * D_ + d] = acc;
  }
}

extern "C" void kernel_launch(void* const* d_in, const int* in_sizes, int n_in,
                              void* d_out, int out_size, void* d_ws, size_t ws_size,
                              hipStream_t stream) {
  const float* x  = (const float*)d_in[0];
  const float* w  = (const float*)d_in[1];
  const float* bh = (const float*)d_in[2];

  float* out          = (float*)d_out;
  float* out_tok      = out;                                        // [B,DENSITY,D]
  float* out_skip     = out_tok  + (size_t)B_ * DENSITY * D_;       // [B,KSKIP,D]
  float* out_sum      = out_skip + (size_t)B_ * KSKIP * D_;         // [B,1,D]
  float* out_sum_skip = out_sum  + (size_t)B_ * D_;                 // [B,1,D]

  // Workspace: logits (B*T f32) | sorted idx (B*T i32) | weights (B*T f32) = 1.5 MB
  float* logits = (float*)d_ws;
  int*   sidx   = (int*)(logits + (size_t)B_ * T_);
  float* wsor   = (float*)(sidx + (size_t)B_ * T_);

  // 1) GEMV via fp32 WMMA: B*T/16 = 8192 wave-tiles, 8 waves per block.
  logits_kernel<<<(B_ * T_ / 16) / 8, 256, 0, stream>>>(x, w, bh, logits);

  // 2) Full descending sort + softmax weights, one block per batch row.
  sort_softmax_kernel<<<B_, 1024, 0, stream>>>(logits, sidx, wsor);

  // 3) Permute all B*T rows into tokens / skip_tokens.
  gather_kernel<<<B_ * T_, 96, 0, stream>>>(x, sidx, out_tok, out_skip);

  // 4) Softmax-weighted summaries.
  summary_kernel<<<dim3(B_, 2), 384, 0, stream>>>(out_tok, out_skip, wsor,
                                                  out_sum, out_sum_skip);
}